// SparseHyperGraphAttentionLayer_81810537055479
// MI455X (gfx1250) — compile-verified
//
#include <hip/hip_runtime.h>

#define IN_F    128
#define OUT_F   128
#define ARITY   8
#define K_EDGES 16
#define ALPHA   0.2f
#define NEG_BIG (-9.0e15f)

typedef __attribute__((ext_vector_type(2))) float v2f;
typedef __attribute__((ext_vector_type(8))) float v8f;

__device__ __forceinline__ float wave_allreduce_sum(float v) {
#pragma unroll
  for (int off = 16; off > 0; off >>= 1)
    v += __shfl_xor(v, off, 32);
  return v;
}

__device__ __forceinline__ float fast_exp(float x) { return __expf(x); }
__device__ __forceinline__ float elu(float x) { return x > 0.0f ? x : fast_exp(x) - 1.0f; }

// ---------------------------------------------------------------------------
// C[M,128] = A[M,128] @ W[128,128] using V_WMMA_F32_16X16X4_F32.
// 256-thread block = 8 waves; wave w owns columns [16w, 16w+16).
// Each wave preloads its B tile (128x16) into 32 v2f registers once, then
// grid-strides over 16-row stripes: 32 WMMA ops per stripe (K = 128).
// A fragment layout (ISA 7.12.2, 32-bit A 16x4): lanes 0-15 hold K={k,k+1},
// lanes 16-31 hold K={k+2,k+3} -> single b64 load per lane per step.
// ---------------------------------------------------------------------------
__global__ void gemm128_wmma(const float* __restrict__ A,
                             const float* __restrict__ W,
                             float* __restrict__ C, int M) {
  const int lane = threadIdx.x & 31;
  const int wave = threadIdx.x >> 5;   // 0..7 -> column block
  const int colBase = wave * 16;
  const int half = lane >> 4;          // 0 or 1
  const int l15 = lane & 15;

  // Preload B tile [128 x 16] (fixed per wave for the whole kernel).
  v2f b[32];
#pragma unroll
  for (int k = 0; k < 32; ++k) {
    const int kk = k * 4 + 2 * half;
    b[k].x = W[(kk + 0) * OUT_F + colBase + l15];
    b[k].y = W[(kk + 1) * OUT_F + colBase + l15];
  }

  const int nstripes = M >> 4;  // M is a multiple of 16 (30000 / 60000)
  for (int s = blockIdx.x; s < nstripes; s += gridDim.x) {
    const int row0 = s * 16;
    const float* Arow = A + (size_t)(row0 + l15) * IN_F + 2 * half;
    v8f c = {};
#pragma unroll
    for (int k = 0; k < 32; ++k) {
      v2f a;
      a.x = Arow[k * 4 + 0];
      a.y = Arow[k * 4 + 1];
      c = __builtin_amdgcn_wmma_f32_16x16x4_f32(
          /*neg_a=*/false, a, /*neg_b=*/false, b[k],
          /*c_mod=*/(short)0, c, /*reuse_a=*/false, /*reuse_b=*/false);
    }
    // C fragment layout: VGPR r -> (M=row0+r+8*half, N=colBase+l15)
#pragma unroll
    for (int r = 0; r < 8; ++r) {
      const int row = row0 + r + 8 * half;
      C[(size_t)row * OUT_F + colBase + l15] = c[r];
    }
  }
}

// ---------------------------------------------------------------------------
// Edge aggregation: one wave per hyperedge. Each lane owns a float4 slice of
// the 128-wide feature rows. Writes pre-ELU new_edge_embs (GEMM input).
// ---------------------------------------------------------------------------
__global__ void edge_agg(const float* __restrict__ Wh,
                         const int* __restrict__ edge_list,
                         const float* __restrict__ a1,
                         float* __restrict__ edge_out, int M) {
  const int lane = threadIdx.x & 31;
  const int wave = threadIdx.x >> 5;
  const int e = blockIdx.x * 8 + wave;
  if (e >= M) return;

  const float4 a1c = *(const float4*)(a1 + lane * 4);

  float4 v[ARITY];
  float sc[ARITY];
#pragma unroll
  for (int r = 0; r < ARITY; ++r) {
    const int idx = edge_list[e * ARITY + r];
    v[r] = *(const float4*)(Wh + (size_t)idx * OUT_F + lane * 4);
    float4 l;
    l.x = v[r].x > 0.0f ? v[r].x : ALPHA * v[r].x;
    l.y = v[r].y > 0.0f ? v[r].y : ALPHA * v[r].y;
    l.z = v[r].z > 0.0f ? v[r].z : ALPHA * v[r].z;
    l.w = v[r].w > 0.0f ? v[r].w : ALPHA * v[r].w;
    float p = l.x * a1c.x + l.y * a1c.y + l.z * a1c.z + l.w * a1c.w;
    p = wave_allreduce_sum(p);
    sc[r] = p > 0.0f ? p : NEG_BIG;
  }

  float mx = sc[0];
#pragma unroll
  for (int r = 1; r < ARITY; ++r) mx = fmaxf(mx, sc[r]);
  float wsum = 0.0f;
  float wgt[ARITY];
#pragma unroll
  for (int r = 0; r < ARITY; ++r) { wgt[r] = fast_exp(sc[r] - mx); wsum += wgt[r]; }
  const float inv = 1.0f / wsum;

  float4 acc = make_float4(0.f, 0.f, 0.f, 0.f);
#pragma unroll
  for (int r = 0; r < ARITY; ++r) {
    const float w = wgt[r] * inv;
    acc.x += w * v[r].x; acc.y += w * v[r].y;
    acc.z += w * v[r].z; acc.w += w * v[r].w;
  }
  *(float4*)(edge_out + (size_t)e * OUT_F + lane * 4) = acc;
}

// ---------------------------------------------------------------------------
// Node aggregation: one wave per node, K=16 incident edges held in registers.
// Applies ELU and writes the final node output.
// ---------------------------------------------------------------------------
__global__ void node_agg(const float* __restrict__ Wf,
                         const float* __restrict__ node_embs,
                         const int* __restrict__ node_list,
                         const float* __restrict__ a2,
                         float* __restrict__ out, int N) {
  const int lane = threadIdx.x & 31;
  const int wave = threadIdx.x >> 5;
  const int n = blockIdx.x * 8 + wave;
  if (n >= N) return;

  const float4 a2lo = *(const float4*)(a2 + lane * 4);
  const float4 a2hi = *(const float4*)(a2 + OUT_F + lane * 4);
  const float4 x = *(const float4*)(node_embs + (size_t)n * IN_F + lane * 4);
  float h2 = x.x * a2hi.x + x.y * a2hi.y + x.z * a2hi.z + x.w * a2hi.w;
  h2 = wave_allreduce_sum(h2);

  float4 v[K_EDGES];
  float sc[K_EDGES];
#pragma unroll
  for (int k = 0; k < K_EDGES; ++k) {
    const int idx = node_list[n * K_EDGES + k];
    v[k] = *(const float4*)(Wf + (size_t)idx * OUT_F + lane * 4);
    float s = v[k].x * a2lo.x + v[k].y * a2lo.y + v[k].z * a2lo.z + v[k].w * a2lo.w;
    s = wave_allreduce_sum(s);
    sc[k] = (s != 0.0f) ? s + h2 : s;
  }

  float mx = sc[0];
#pragma unroll
  for (int k = 1; k < K_EDGES; ++k) mx = fmaxf(mx, sc[k]);
  float wsum = 0.0f;
  float wgt[K_EDGES];
#pragma unroll
  for (int k = 0; k < K_EDGES; ++k) { wgt[k] = fast_exp(sc[k] - mx); wsum += wgt[k]; }
  const float inv = 1.0f / wsum;

  float4 acc = make_float4(0.f, 0.f, 0.f, 0.f);
#pragma unroll
  for (int k = 0; k < K_EDGES; ++k) {
    const float w = wgt[k] * inv;
    acc.x += w * v[k].x; acc.y += w * v[k].y;
    acc.z += w * v[k].z; acc.w += w * v[k].w;
  }
  acc.x = elu(acc.x); acc.y = elu(acc.y); acc.z = elu(acc.z); acc.w = elu(acc.w);
  *(float4*)(out + (size_t)n * OUT_F + lane * 4) = acc;
}

// In-place ELU over the edge-output segment (GEMM already read pre-ELU data).
__global__ void elu_inplace(float* __restrict__ p, int n) {
  const int i = blockIdx.x * blockDim.x + threadIdx.x;
  if (i < n) p[i] = elu(p[i]);
}

extern "C" void kernel_launch(void* const* d_in, const int* in_sizes, int n_in,
                              void* d_out, int out_size, void* d_ws, size_t ws_size,
                              hipStream_t stream) {
  const float* node_embs = (const float*)d_in[0];
  // d_in[1] (edge_embs) is unused by the reference computation.
  const int* edge_list = (const int*)d_in[2];
  const int* node_list = (const int*)d_in[3];
  const float* W1 = (const float*)d_in[4];
  const float* W2 = (const float*)d_in[5];
  const float* a1 = (const float*)d_in[6];
  const float* a2 = (const float*)d_in[7];

  const int N = in_sizes[0] / IN_F;    // 30000
  const int M = in_sizes[2] / ARITY;   // 60000

  float* Wh = (float*)d_ws;                          // N*128 f32 (15.4 MB)
  float* Wf = Wh + (size_t)N * OUT_F;                // M*128 f32 (30.7 MB)
  float* out_nodes = (float*)d_out;                  // N*128
  float* out_edges = out_nodes + (size_t)N * OUT_F;  // M*128 (pre-ELU, then ELU)

  // 1) Wh = node_embs @ W1
  gemm128_wmma<<<dim3(512), dim3(256), 0, stream>>>(node_embs, W1, Wh, N);
  // 2) edge attention -> pre-ELU new_edge_embs straight into d_out edge slot
  edge_agg<<<dim3((M + 7) / 8), dim3(256), 0, stream>>>(Wh, edge_list, a1, out_edges, M);
  // 3) Wf = new_edge_embs @ W2
  gemm128_wmma<<<dim3(512), dim3(256), 0, stream>>>(out_edges, W2, Wf, M);
  // 4) node attention + ELU -> d_out node slot
  node_agg<<<dim3((N + 7) / 8), dim3(256), 0, stream>>>(Wf, node_embs, node_list, a2, out_nodes, N);
  // 5) ELU in place on edge outputs
  elu_inplace<<<dim3((M * OUT_F + 255) / 256), dim3(256), 0, stream>>>(out_edges, M * OUT_F);
}